// KLDivergence_5927054868988
// MI455X (gfx1250) — compile-verified
//
#include <hip/hip_runtime.h>
#include <stdint.h>

// KL(true||pred) with per-batch normalization, B=64, N=512*512.
// Single streaming pass: per batch accumulate T=sum(t+eps), P=sum(p+eps),
// S=sum((t+eps)*(log(t+eps)-log(p+eps))); kl_b = S/T + log(P/T); out = mean.
//
// CDNA5 path: global_load_async_to_lds_b128 (ASYNCcnt) double-buffered per-wave
// LDS staging + s_wait_asynccnt, ds_load_b128 reads, wave32 shfl reduction.

#define KL_EPS 1e-6f
#define THREADS 256
#define WAVES 8
#define BATCH 64
#define F4_PER_BATCH 65536        // 262144 floats / 4
#define STAGES 8                  // float4 per thread per block
#define BLOCKS_PER_BATCH 32       // 65536 / (STAGES*THREADS)

// global_load_async_to_lds_b128 vdst(lds byte addr), vaddr(u32 byte offset), saddr(base)
#define ASYNC_B128(ldsoff, voff, base)                                   \
  asm volatile("global_load_async_to_lds_b128 %0, %1, %2"                \
               :: "v"(ldsoff), "v"(voff), "s"(base) : "memory")

__global__ __launch_bounds__(THREADS) void kl_zero(float* ws) {
  ws[threadIdx.x] = 0.0f;   // 256 floats >= 64 batches * 4 slots
}

__global__ __launch_bounds__(THREADS) void kl_main(const float* __restrict__ pred,
                                                   const float* __restrict__ truem,
                                                   float* __restrict__ ws) {
  // [wave][buf(2)][arr(2)][lane(32)] * 16B = 16 KB
  __shared__ __align__(16) unsigned char smem[WAVES * 2 * 2 * 32 * 16];

  const int tid  = threadIdx.x;
  const int lane = tid & 31;
  const int wave = tid >> 5;
  const int b    = blockIdx.y;

  // This thread's stage-0 float4 index within the whole pred/true arrays.
  const uint32_t f4base =
      (uint32_t)b * F4_PER_BATCH + (uint32_t)blockIdx.x * (STAGES * THREADS) + (uint32_t)tid;

  const uint64_t predBase = (uint64_t)(uintptr_t)pred;
  const uint64_t trueBase = (uint64_t)(uintptr_t)truem;

  // Per-wave private LDS slice: no cross-wave sharing -> no barriers needed.
  unsigned char* wbase = &smem[wave * 2048 + lane * 16];
  const float4* rdP[2] = { (const float4*)(wbase +    0), (const float4*)(wbase + 1024) };
  const float4* rdT[2] = { (const float4*)(wbase +  512), (const float4*)(wbase + 1536) };
  // Low 32 bits of a generic pointer to LDS == LDS byte offset (LDS aperture).
  const uint32_t ldsP[2] = { (uint32_t)(uintptr_t)rdP[0], (uint32_t)(uintptr_t)rdP[1] };
  const uint32_t ldsT[2] = { (uint32_t)(uintptr_t)rdT[0], (uint32_t)(uintptr_t)rdT[1] };

  // Prologue: stage 0 into buffer 0 (ASYNCcnt += 2).
  {
    const uint32_t go = f4base * 16u;
    ASYNC_B128(ldsP[0], go, predBase);
    ASYNC_B128(ldsT[0], go, trueBase);
  }

  float sT = 0.0f, sP = 0.0f, sS = 0.0f;

  #pragma unroll
  for (int s = 0; s < STAGES; ++s) {
    const int buf = s & 1;
    if (s + 1 < STAGES) {
      // Prefetch next stage into the other buffer, then wait for current stage
      // (async loads complete in order; 2 newest remain outstanding).
      const uint32_t go = (f4base + (uint32_t)(s + 1) * THREADS) * 16u;
      ASYNC_B128(ldsP[buf ^ 1], go, predBase);
      ASYNC_B128(ldsT[buf ^ 1], go, trueBase);
      asm volatile("s_wait_asynccnt 2" ::: "memory");
    } else {
      asm volatile("s_wait_asynccnt 0" ::: "memory");
    }

    const float4 p = *rdP[buf];   // ds_load_b128
    const float4 t = *rdT[buf];

    float tp, pp;
    tp = t.x + KL_EPS; pp = p.x + KL_EPS;
    sT += tp; sP += pp; sS += tp * (__logf(tp) - __logf(pp));
    tp = t.y + KL_EPS; pp = p.y + KL_EPS;
    sT += tp; sP += pp; sS += tp * (__logf(tp) - __logf(pp));
    tp = t.z + KL_EPS; pp = p.z + KL_EPS;
    sT += tp; sP += pp; sS += tp * (__logf(tp) - __logf(pp));
    tp = t.w + KL_EPS; pp = p.w + KL_EPS;
    sT += tp; sP += pp; sS += tp * (__logf(tp) - __logf(pp));
  }

  // wave32 butterfly reduction
  #pragma unroll
  for (int o = 16; o > 0; o >>= 1) {
    sT += __shfl_xor(sT, o, 32);
    sP += __shfl_xor(sP, o, 32);
    sS += __shfl_xor(sS, o, 32);
  }

  if (lane == 0) {
    atomicAdd(&ws[b * 4 + 0], sT);
    atomicAdd(&ws[b * 4 + 1], sP);
    atomicAdd(&ws[b * 4 + 2], sS);
  }
}

__global__ __launch_bounds__(64) void kl_final(const float* __restrict__ ws,
                                               float* __restrict__ out) {
  const int b = threadIdx.x;           // 64 threads = 2 waves
  const float T = ws[b * 4 + 0];
  const float P = ws[b * 4 + 1];
  const float S = ws[b * 4 + 2];
  float kl = S / T + __logf(P / T);

  #pragma unroll
  for (int o = 16; o > 0; o >>= 1) kl += __shfl_xor(kl, o, 32);

  __shared__ float tmp[2];
  if ((threadIdx.x & 31) == 0) tmp[threadIdx.x >> 5] = kl;
  __syncthreads();
  if (threadIdx.x == 0) out[0] = (tmp[0] + tmp[1]) * (1.0f / 64.0f);
}

extern "C" void kernel_launch(void* const* d_in, const int* in_sizes, int n_in,
                              void* d_out, int out_size, void* d_ws, size_t ws_size,
                              hipStream_t stream) {
  const float* pred = (const float*)d_in[0];
  const float* tru  = (const float*)d_in[1];
  float* ws  = (float*)d_ws;     // needs 256 floats = 1 KB
  float* out = (float*)d_out;

  hipLaunchKernelGGL(kl_zero,  dim3(1), dim3(THREADS), 0, stream, ws);
  hipLaunchKernelGGL(kl_main,  dim3(BLOCKS_PER_BATCH, BATCH), dim3(THREADS), 0, stream,
                     pred, tru, ws);
  hipLaunchKernelGGL(kl_final, dim3(1), dim3(64), 0, stream, ws, out);
}